// CapsuleLayer_59201829208658
// MI455X (gfx1250) — compile-verified
//
#include <hip/hip_runtime.h>

// CapsuleLayer dynamic routing on gfx1250 (MI455X), fp32 WMMA (16x16x4).
// Strategy: never materialize u_hat (512MB); recompute per pass via WMMA.
// W (128MB) + x/xT (16MB) + routing state (~32MB) all stay L2-resident (192MB).
// All hot-loop operand streams are lane-coalesced:
//   - A operand from xT2[i][kpair][b] (float2 k-pairs, batch contiguous)
//   - routing logits/weights kept transposed as [j][i][b]
//   - W[j][i] blocks read once each, prefetched 2 iterations ahead
//   - bupdate row-dot done via conflict-free LDS transpose (no bpermute chains)

typedef float v2f __attribute__((ext_vector_type(2)));
typedef float v8f __attribute__((ext_vector_type(8)));

#define B_SZ 64
#define I_SZ 2048
#define DIN  16
#define J_SZ 32
#define D_SZ 32
#define PT_PITCH 18   // LDS row pitch (dwords): disjoint bank ranges for lane halves

// ---------------------------------------------------------------- zero init
__global__ void zero_kernel(float* __restrict__ p, int n) {
    int t = blockIdx.x * blockDim.x + threadIdx.x;
    int stride = gridDim.x * blockDim.x;
    for (int k = t; k < n; k += stride) p[k] = 0.0f;
}

// ------------------------------------------------------- x transpose (once)
// x[b][i][k] -> xT2[i][k/2][b] as float2 (k-pair innermost per b slot).
__global__ void xpose_kernel(const float* __restrict__ x, float* __restrict__ xT2) {
    int t = blockIdx.x * blockDim.x + threadIdx.x;   // 0 .. B*I*DIN
    if (t >= B_SZ * I_SZ * DIN) return;
    int b = t >> 15;            // / (I_SZ*DIN)
    int rem = t & 32767;
    int i = rem >> 4;
    int k = rem & 15;
    xT2[(((size_t)i * 8 + (k >> 1)) * B_SZ + b) * 2 + (k & 1)] = x[t];
}

// ---------------------------------------------------------------- wsum pass
// s[b,j,d] += sum_i c[b,j,i] * u_hat[b,j,i,d], u_hat recomputed via WMMA.
// Per (j,i): C[64x32] = (c .* rows of X_i)[64x16] @ W_{j,i}^T[16x32].
// grid = (I/64, J), block = 256 (8 waves), each wave handles 8 consecutive i.
template <bool HAS_CW>
__global__ __launch_bounds__(256)
void wsum_kernel(const float* __restrict__ xT2, const float* __restrict__ W,
                 const float* __restrict__ cT,   // [j][i][b], used if HAS_CW
                 float wscale,                   // folded at write-out
                 float* __restrict__ s_out)      // [b][j][d]
{
    const int j    = blockIdx.y;
    const int wave = threadIdx.x >> 5;
    const int lane = threadIdx.x & 31;
    const int lh   = lane >> 4;     // lane half (K split / M split)
    const int ln   = lane & 15;
    const int i0   = blockIdx.x * 64 + wave * 8;
    const v2f* xt  = (const v2f*)xT2;

    v8f acc[4][2] = {};             // [m-tile of b][n-tile of d]

    for (int ii = 0; ii < 8; ++ii) {
        const int i = i0 + ii;
        const float* wp = W + ((size_t)j * I_SZ + i) * (D_SZ * DIN);
        // prefetch the W block 2 iterations ahead: 32 lanes x 64B = 2KB
        __builtin_prefetch(wp + 2 * D_SZ * DIN + lane * 16, 0, 3);
        // B operand: B[k,n] = W[j,i, d=n-tile+ln, c=k]
        v2f bf[2][4];
#pragma unroll
        for (int n = 0; n < 2; ++n)
#pragma unroll
            for (int ks = 0; ks < 4; ++ks)
                bf[n][ks] = *(const v2f*)(wp + (n * 16 + ln) * DIN + ks * 4 + 2 * lh);
#pragma unroll
        for (int m = 0; m < 4; ++m) {
            const int brow = m * 16 + ln;   // batch row for this A fragment
            v2f af[4];
#pragma unroll
            for (int ks = 0; ks < 4; ++ks)   // coalesced b64: k-pair (ks*2+lh)
                af[ks] = xt[((size_t)i * 8 + ks * 2 + lh) * B_SZ + brow];
            if (HAS_CW) {
                const float cv = cT[((size_t)j * I_SZ + i) * B_SZ + brow];
#pragma unroll
                for (int ks = 0; ks < 4; ++ks) { af[ks].x *= cv; af[ks].y *= cv; }
            }
#pragma unroll
            for (int n = 0; n < 2; ++n)
#pragma unroll
                for (int ks = 0; ks < 4; ++ks)
                    acc[m][n] = __builtin_amdgcn_wmma_f32_16x16x4_f32(
                        false, af[ks], false, bf[n][ks], (short)0,
                        acc[m][n], false, false);
        }
    }

    // Reduce the 8 waves' partial [64x32] tiles in LDS, then one atomic pass.
    __shared__ float stile[B_SZ * D_SZ];
    for (int t = threadIdx.x; t < B_SZ * D_SZ; t += 256) stile[t] = 0.0f;
    __syncthreads();
#pragma unroll
    for (int m = 0; m < 4; ++m)
#pragma unroll
        for (int n = 0; n < 2; ++n)
#pragma unroll
            for (int r = 0; r < 8; ++r) {
                int b = m * 16 + r + 8 * lh;      // C-layout: vgpr r -> M=r / M=r+8
                int d = n * 16 + ln;
                atomicAdd(&stile[b * D_SZ + d], acc[m][n][r]);
            }
    __syncthreads();
    for (int t = threadIdx.x; t < B_SZ * D_SZ; t += 256) {
        int b = t >> 5, d = t & 31;
        atomicAdd(s_out + ((size_t)b * J_SZ + j) * D_SZ + d, stile[t] * wscale);
    }
}

// -------------------------------------------------------------- logit update
// blogT[j,i,b] += sum_k x[b,i,k] * h[b,k],  h = V_j[64x32] @ W_{j,i}[32x16]
// (== sum_d v[b,j,d] * u_hat[b,j,i,d]).  Same grid shape as wsum.
// Row-dot via per-wave LDS transpose: conflict-free stores (pitch 18), then
// each lane serially sums 2 full rows and does a coalesced blogT update.
__global__ __launch_bounds__(256)
void bupdate_kernel(const float* __restrict__ x, const float* __restrict__ W,
                    const float* __restrict__ v, float* __restrict__ blogT)
{
    const int j    = blockIdx.y;
    const int wave = threadIdx.x >> 5;
    const int lane = threadIdx.x & 31;
    const int lh   = lane >> 4;
    const int ln   = lane & 15;
    const int i0   = blockIdx.x * 64 + wave * 8;

    __shared__ float pt[8 * B_SZ * PT_PITCH];   // 36 KB, per-wave regions
    float* myp = pt + wave * (B_SZ * PT_PITCH);

    // A = v[b, j, d] : i-invariant, load once. [4 m-tiles][8 K-steps over d]
    v2f af[4][8];
#pragma unroll
    for (int m = 0; m < 4; ++m)
#pragma unroll
        for (int ks = 0; ks < 8; ++ks)
            af[m][ks] = *(const v2f*)(v + ((size_t)(m * 16 + ln) * J_SZ + j) * D_SZ
                                        + ks * 4 + 2 * lh);

    for (int ii = 0; ii < 8; ++ii) {
        const int i = i0 + ii;
        const float* wp = W + ((size_t)j * I_SZ + i) * (D_SZ * DIN);
        __builtin_prefetch(wp + 2 * D_SZ * DIN + lane * 16, 0, 3);
        v2f bf[8];                       // B[kdim=d, n=k] = W[j,i,d, k=ln]
#pragma unroll
        for (int ks = 0; ks < 8; ++ks) {
            int d0 = ks * 4 + 2 * lh;
            bf[ks].x = wp[(size_t)d0 * DIN + ln];
            bf[ks].y = wp[(size_t)(d0 + 1) * DIN + ln];
        }
        v8f acc[4] = {};
#pragma unroll
        for (int m = 0; m < 4; ++m)
#pragma unroll
            for (int ks = 0; ks < 8; ++ks)
                acc[m] = __builtin_amdgcn_wmma_f32_16x16x4_f32(
                    false, af[m][ks], false, bf[ks], (short)0, acc[m], false, false);

        // products p[b,k] = h[b,k] * x[b,i,k], stored transposed into LDS
#pragma unroll
        for (int m = 0; m < 4; ++m)
#pragma unroll
            for (int r = 0; r < 8; ++r) {
                int brow = m * 16 + r + 8 * lh;   // C-layout row for this vgpr
                float p = acc[m][r] * x[((size_t)brow * I_SZ + i) * DIN + ln];
                myp[brow * PT_PITCH + ln] = p;
            }
        // each lane owns 2 full rows: serial 16-element sum + coalesced update
#pragma unroll
        for (int rr = 0; rr < 2; ++rr) {
            int brow = lane + rr * 32;
            const v2f* rowp = (const v2f*)(myp + brow * PT_PITCH);
            float sum = 0.0f;
#pragma unroll
            for (int q = 0; q < 8; ++q) { v2f t = rowp[q]; sum += t.x + t.y; }
            size_t off = ((size_t)j * I_SZ + i) * B_SZ + brow;
            blogT[off] += sum;
        }
    }
}

// ------------------------------------------------------------ softmax over j
// blogT/cT layout [j][i][b]: thread t = i*64 + b -> fully coalesced.
__global__ void softmax_kernel(const float* __restrict__ blogT, float* __restrict__ cT)
{
    int t = blockIdx.x * blockDim.x + threadIdx.x;
    if (t >= B_SZ * I_SZ) return;
    int b = t & 63;
    int i = t >> 6;
    float vals[J_SZ];
    float mx = -3.4e38f;
#pragma unroll
    for (int jj = 0; jj < J_SZ; ++jj) {
        vals[jj] = blogT[((size_t)jj * I_SZ + i) * B_SZ + b];
        mx = fmaxf(mx, vals[jj]);
    }
    float sum = 0.0f;
#pragma unroll
    for (int jj = 0; jj < J_SZ; ++jj) {
        float e = __expf(vals[jj] - mx);
        vals[jj] = e;
        sum += e;
    }
    float inv = 1.0f / sum;
#pragma unroll
    for (int jj = 0; jj < J_SZ; ++jj)
        cT[((size_t)jj * I_SZ + i) * B_SZ + b] = vals[jj] * inv;
}

// ---------------------------------------------------------------- squash
__global__ void squash_kernel(const float* __restrict__ s, float* __restrict__ out, int n)
{
    int t = blockIdx.x * blockDim.x + threadIdx.x;   // t indexes (b, j)
    if (t >= n) return;
    const float* p = s + (size_t)t * D_SZ;
    float ss = 0.0f;
#pragma unroll
    for (int d = 0; d < D_SZ; ++d) ss += p[d] * p[d];
    float scale = ss / (1.0f + ss) / sqrtf(ss + 1e-7f);
    float* q = out + (size_t)t * D_SZ;
#pragma unroll
    for (int d = 0; d < D_SZ; ++d) q[d] = p[d] * scale;
}

// ---------------------------------------------------------------- launcher
extern "C" void kernel_launch(void* const* d_in, const int* in_sizes, int n_in,
                              void* d_out, int out_size, void* d_ws, size_t ws_size,
                              hipStream_t stream)
{
    (void)in_sizes; (void)n_in; (void)out_size; (void)ws_size;
    const float* x = (const float*)d_in[0];   // [64, 2048, 16]
    const float* W = (const float*)d_in[1];   // [32, 2048, 32, 16]
    float* out = (float*)d_out;               // [64, 32, 32]

    // ws layout (floats): blogT | s0 | s1 | s2 | v0 | v1 | cT | xT2  (~43 MB)
    float* ws = (float*)d_ws;
    const size_t N_BLOG = (size_t)B_SZ * J_SZ * I_SZ;   // 4,194,304
    const size_t N_S    = (size_t)B_SZ * J_SZ * D_SZ;   //    65,536
    const size_t N_X    = (size_t)B_SZ * I_SZ * DIN;    // 2,097,152
    float* blogT = ws;
    float* s0    = blogT + N_BLOG;
    float* s1    = s0 + N_S;
    float* s2    = s1 + N_S;
    float* v0    = s2 + N_S;
    float* v1    = v0 + N_S;
    float* cT    = v1 + N_S;
    float* xT2   = cT + N_BLOG;

    // zero accumulators (blogT + s0..s2 are contiguous), transpose x once
    zero_kernel<<<2048, 256, 0, stream>>>(blogT, (int)(N_BLOG + 3 * N_S));
    xpose_kernel<<<(int)(N_X / 256), 256, 0, stream>>>(x, xT2);

    dim3 gemm_grid(I_SZ / 64, J_SZ);   // (32, 32) workgroups of 8 waves

    // routing iter 0: c uniform = 1/J (folded into write-out scale)
    wsum_kernel<false><<<gemm_grid, 256, 0, stream>>>(xT2, W, nullptr, 1.0f / J_SZ, s0);
    squash_kernel<<<8, 256, 0, stream>>>(s0, v0, B_SZ * J_SZ);
    bupdate_kernel<<<gemm_grid, 256, 0, stream>>>(x, W, v0, blogT);

    // routing iter 1
    softmax_kernel<<<(B_SZ * I_SZ) / 256, 256, 0, stream>>>(blogT, cT);
    wsum_kernel<true><<<gemm_grid, 256, 0, stream>>>(xT2, W, cT, 1.0f, s1);
    squash_kernel<<<8, 256, 0, stream>>>(s1, v1, B_SZ * J_SZ);
    bupdate_kernel<<<gemm_grid, 256, 0, stream>>>(x, W, v1, blogT);

    // routing iter 2 (final)
    softmax_kernel<<<(B_SZ * I_SZ) / 256, 256, 0, stream>>>(blogT, cT);
    wsum_kernel<true><<<gemm_grid, 256, 0, stream>>>(xT2, W, cT, 1.0f, s2);
    squash_kernel<<<8, 256, 0, stream>>>(s2, out, B_SZ * J_SZ);
}